// GNNClassifier_45741401702804
// MI455X (gfx1250) — compile-verified
//
#include <hip/hip_runtime.h>

#define DH 128
#define POOL_ROWS 250

typedef __attribute__((ext_vector_type(2))) float v2f;
typedef __attribute__((ext_vector_type(8))) float v8f;

// Guaranteed HW fp32 atomic at device scope (L2 atomic units), no CAS loop.
__device__ __forceinline__ void atomAddF32(float* p, float v) {
  asm volatile("global_atomic_add_f32 %0, %1, off scope:SCOPE_DEV"
               :: "v"(p), "v"(v) : "memory");
}

// deg = 1.0 (self loop contribution), agg buffer = 0, gsum = 0
__global__ void k_init(float* __restrict__ deg, float* __restrict__ agg,
                       float* __restrict__ gsum, int n) {
  const int idx = blockIdx.x * blockDim.x + threadIdx.x;
  if (idx < n * DH) agg[idx] = 0.0f;
  if (idx < n)      deg[idx] = 1.0f;
  if (idx < DH)     gsum[idx] = 0.0f;
}

__global__ void k_zero(float* __restrict__ p, int total) {
  const int idx = blockIdx.x * blockDim.x + threadIdx.x;
  if (idx < total) p[idx] = 0.0f;
}

__global__ void k_degree(float* __restrict__ deg, const int* __restrict__ col, int ne) {
  const int e = blockIdx.x * blockDim.x + threadIdx.x;
  if (e < ne) atomAddF32(&deg[col[e]], 1.0f);
}

__global__ void k_rsqrt(float* __restrict__ deg, int n) {
  const int i = blockIdx.x * blockDim.x + threadIdx.x;
  if (i < n) {
    const float d = deg[i];
    deg[i] = (d > 0.0f) ? rsqrtf(d) : 0.0f;
  }
}

// H[nrows,128] = X[nrows,128] @ W[128,128] using V_WMMA_F32_16X16X4_F32.
// One wave computes a 16-row x 128-col strip; 8 waves (256 thr) per block.
__global__ __launch_bounds__(256) void k_gemm_wmma(const float* __restrict__ X,
                                                   const float* __restrict__ W,
                                                   float* __restrict__ H, int nrows) {
  __shared__ float Ws[DH * DH];  // 64 KB
  for (int i = threadIdx.x; i < DH * DH; i += 256) Ws[i] = W[i];
  __syncthreads();

  const int lane = threadIdx.x & 31;
  const int wave = threadIdx.x >> 5;
  const int rowBase = (blockIdx.x * 8 + wave) * 16;
  if (rowBase >= nrows) return;  // wave-uniform; EXEC stays all-1s for WMMA

  const int m  = lane & 15;          // row (A) / col (B/D) within tile
  const int kh = (lane >> 4) << 1;   // K sub-offset: 0 for lanes 0-15, 2 for 16-31

  int srcRow = rowBase + m;
  if (srcRow >= nrows) srcRow = nrows - 1;  // clamp load, keep EXEC full
  const float* __restrict__ xr = X + (size_t)srcRow * DH;

  // A fragments for all K=128 (32 steps of K=4), register resident
  v2f a[32];
#pragma unroll
  for (int kk = 0; kk < 32; ++kk) {
    const int k0 = kk * 4 + kh;
    v2f t = {xr[k0], xr[k0 + 1]};
    a[kk] = t;
  }

  const int mBase = rowBase + ((lane >> 4) << 3);  // D rows covered by this lane's acc
  for (int ct = 0; ct < 8; ++ct) {
    const int colBase = ct * 16;
    v8f acc = {0.f, 0.f, 0.f, 0.f, 0.f, 0.f, 0.f, 0.f};
#pragma unroll
    for (int kk = 0; kk < 32; ++kk) {
      const int k0 = kk * 4 + kh;
      v2f b = {Ws[k0 * DH + colBase + m], Ws[(k0 + 1) * DH + colBase + m]};
      acc = __builtin_amdgcn_wmma_f32_16x16x4_f32(false, a[kk], false, b,
                                                  (short)0, acc, false, false);
    }
#pragma unroll
    for (int v = 0; v < 8; ++v) {
      const int r = mBase + v;
      if (r < nrows) H[(size_t)r * DH + colBase + m] = acc[v];
    }
  }
}

// One wave per edge: AGG[col] += dinv[row]*dinv[col] * H[row]
// Each lane owns 4 consecutive features: one b128 load + 4 fp32 atomics.
__global__ __launch_bounds__(256) void k_aggregate(const float* __restrict__ H,
                                                   const float* __restrict__ dinv,
                                                   const int* __restrict__ row,
                                                   const int* __restrict__ col,
                                                   float* __restrict__ AGG, int ne) {
  const int lane = threadIdx.x & 31;
  const int e = (int)((blockIdx.x * blockDim.x + threadIdx.x) >> 5);
  if (e >= ne) return;
  const int r = row[e];
  const int c = col[e];
  const float w = dinv[r] * dinv[c];
  const float4* __restrict__ hr4 = (const float4*)(H + (size_t)r * DH);
  const float4 hv = hr4[lane];                     // global_load_b128
  float* __restrict__ ac = AGG + (size_t)c * DH + lane * 4;
  atomAddF32(ac + 0, w * hv.x);
  atomAddF32(ac + 1, w * hv.y);
  atomAddF32(ac + 2, w * hv.z);
  atomAddF32(ac + 3, w * hv.w);
}

// AGG = relu(AGG + dinv^2 * H (self loop) + b), in place
__global__ void k_bias_relu(float* __restrict__ AGG, const float* __restrict__ H,
                            const float* __restrict__ dinv,
                            const float* __restrict__ b, int n) {
  const int idx = blockIdx.x * blockDim.x + threadIdx.x;
  if (idx >= n * DH) return;
  const int i = idx >> 7;
  const int d = idx & (DH - 1);
  const float di = dinv[i];
  const float v = AGG[idx] + di * di * H[idx] + b[d];
  AGG[idx] = v > 0.0f ? v : 0.0f;
}

__global__ void k_pool(const float* __restrict__ H, float* __restrict__ gsum, int n) {
  const int d = threadIdx.x;  // 128 threads
  const int r0 = blockIdx.x * POOL_ROWS;
  int r1 = r0 + POOL_ROWS;
  if (r1 > n) r1 = n;
  float s = 0.0f;
  for (int r = r0; r < r1; ++r) s += H[(size_t)r * DH + d];
  atomAddF32(&gsum[d], s);
}

__global__ void k_classify(const float* __restrict__ gsum, const float* __restrict__ Wc,
                           const float* __restrict__ bc, float* __restrict__ out,
                           float invN, int dout) {
  const int j = threadIdx.x;
  if (j >= dout) return;
  float s = bc[j];
  for (int k = 0; k < DH; ++k) s += (gsum[k] * invN) * Wc[k * dout + j];
  out[j] = s;
}

extern "C" void kernel_launch(void* const* d_in, const int* in_sizes, int n_in,
                              void* d_out, int out_size, void* d_ws, size_t ws_size,
                              hipStream_t stream) {
  const float* x  = (const float*)d_in[0];
  const int*   ei = (const int*)  d_in[1];
  const float* W1 = (const float*)d_in[2];
  const float* b1 = (const float*)d_in[3];
  const float* W2 = (const float*)d_in[4];
  const float* b2 = (const float*)d_in[5];
  const float* Wc = (const float*)d_in[6];
  const float* bc = (const float*)d_in[7];
  float* out = (float*)d_out;

  const int N = in_sizes[0] / DH;   // 50000
  const int E = in_sizes[1] / 2;    // 800000
  const int DOUT = out_size;        // 40
  const int* row = ei;              // edge_index[0]
  const int* col = ei + E;          // edge_index[1]

  // workspace layout (fp32): deg/dinv | bufA (GEMM out) | bufB (agg) | gsum
  float* deg  = (float*)d_ws;
  float* bufA = deg + (((size_t)N + 255) & ~(size_t)255);
  float* bufB = bufA + (size_t)N * DH;
  float* gsum = bufB + (size_t)N * DH;

  const int totNF = N * DH;
  const int gElem = (totNF + 255) / 256;
  const int gEdge = (E + 255) / 256;
  const int gNode = (N + 255) / 256;
  const int gGemm = ((N + 15) / 16 + 7) / 8;   // 8 row-tiles per block
  const int gAgg  = (E + 7) / 8;               // 8 edges (waves) per block
  const int gPool = (N + POOL_ROWS - 1) / POOL_ROWS;

  k_init  <<<gElem, 256, 0, stream>>>(deg, bufB, gsum, N);
  k_degree<<<gEdge, 256, 0, stream>>>(deg, col, E);
  k_rsqrt <<<gNode, 256, 0, stream>>>(deg, N);

  // layer 1: h1 = X@W1 -> scatter -> +selfloop+bias, relu (result in bufB)
  k_gemm_wmma<<<gGemm, 256, 0, stream>>>(x, W1, bufA, N);
  k_aggregate<<<gAgg, 256, 0, stream>>>(bufA, deg, row, col, bufB, E);
  k_bias_relu<<<gElem, 256, 0, stream>>>(bufB, bufA, deg, b1, N);

  // layer 2: h2 = h1@W2 -> scatter -> +selfloop+bias, relu (result in bufB)
  k_gemm_wmma<<<gGemm, 256, 0, stream>>>(bufB, W2, bufA, N);
  k_zero     <<<gElem, 256, 0, stream>>>(bufB, totNF);
  k_aggregate<<<gAgg, 256, 0, stream>>>(bufA, deg, row, col, bufB, E);
  k_bias_relu<<<gElem, 256, 0, stream>>>(bufB, bufA, deg, b2, N);

  // mean pool + classifier
  k_pool    <<<gPool, DH, 0, stream>>>(bufB, gsum, N);
  k_classify<<<1, 64, 0, stream>>>(gsum, Wc, bc, out, 1.0f / (float)N, DOUT);
}